// RNNDetector_13108240188035
// MI455X (gfx1250) — compile-verified
//
#include <hip/hip_runtime.h>
#include <hip/hip_bf16.h>

typedef __attribute__((ext_vector_type(2))) float v2f;
typedef __attribute__((ext_vector_type(8))) float v8f;

// ---------------------------------------------------------------------------
// Fast activations (lower to v_exp_f32 transcendentals, saturate cleanly)
// ---------------------------------------------------------------------------
__device__ __forceinline__ float fast_sigmoid(float x) {
    return 1.0f / (1.0f + __expf(-x));
}
__device__ __forceinline__ float fast_tanh(float x) {
    // 1 - 2/(e^{2x}+1): saturates to +/-1 without inf/inf NaNs
    return 1.0f - 2.0f / (__expf(2.0f * x) + 1.0f);
}

// ---------------------------------------------------------------------------
// Kernel 1: pre0[t][j] = sum_i seq(t,i)*W_ih0[j][i] + b_ih0[j] + b_hh0[j]
// seq(t,i) = (t+i<5) ? 1 : y[t+i-5].  One block per timestep, 256 threads.
// Bandwidth bound (268MB write -> ~11.5us at 23.3 TB/s).
// ---------------------------------------------------------------------------
__global__ __launch_bounds__(256) void pre0_kernel(
    const float* __restrict__ y, const float* __restrict__ W_ih0,
    const float* __restrict__ b_ih0, const float* __restrict__ b_hh0,
    float* __restrict__ pre, int S)
{
    int t = blockIdx.x;
    int j = threadIdx.x;
    if (t >= S) return;
    float x[6];
#pragma unroll
    for (int i = 0; i < 6; ++i) {
        int idx = t + i - 5;
        x[i] = (idx < 0) ? 1.0f : y[idx];
    }
    const float* w = W_ih0 + (size_t)j * 6;
    float acc = b_ih0[j] + b_hh0[j];
#pragma unroll
    for (int i = 0; i < 6; ++i) acc = fmaf(x[i], w[i], acc);
    pre[(size_t)t * 256 + j] = acc;
}

// ---------------------------------------------------------------------------
// Kernel 2/4: serial LSTM scan for one layer (latency-bound critical path).
// pre: [S,256] (input contribution incl. both biases), W_hh: [256,64] row-major.
// One workgroup, 256 threads (8 wave32). Thread j owns gate row j:
//   - W_hh[j][0..63] preloaded into 64 VGPRs
//   - h[64] lives in LDS (broadcast reads), c[j] lives in thread j's register
// Two barriers per step; next step's pre row prefetched during gate math.
// ---------------------------------------------------------------------------
__global__ __launch_bounds__(256) void lstm_scan_kernel(
    const float* __restrict__ pre, const float* __restrict__ W_hh,
    float* __restrict__ htrace, int S)
{
    __shared__ __align__(16) float h_sh[64];
    __shared__ float g_sh[256];
    const int j = threadIdx.x;

    // Preload recurrent weight row into VGPRs
    float w[64];
    const float4* wrow = (const float4*)(W_hh + (size_t)j * 64);
#pragma unroll
    for (int k = 0; k < 16; ++k) {
        float4 v = wrow[k];
        w[4 * k + 0] = v.x; w[4 * k + 1] = v.y;
        w[4 * k + 2] = v.z; w[4 * k + 3] = v.w;
    }

    float c = 0.0f;
    if (j < 64) h_sh[j] = 0.0f;
    __syncthreads();

    float p = pre[j];  // prefetched input contribution for t=0
    for (int t = 0; t < S; ++t) {
        // g[j] = p + dot(W_hh[j,:], h)  -- 4 accumulators break the FMA chain
        float a0 = 0.f, a1 = 0.f, a2 = 0.f, a3 = 0.f;
        const float4* h4 = (const float4*)h_sh;
#pragma unroll
        for (int k = 0; k < 16; ++k) {
            float4 hv = h4[k];
            a0 = fmaf(w[4 * k + 0], hv.x, a0);
            a1 = fmaf(w[4 * k + 1], hv.y, a1);
            a2 = fmaf(w[4 * k + 2], hv.z, a2);
            a3 = fmaf(w[4 * k + 3], hv.w, a3);
        }
        float g = p + ((a0 + a1) + (a2 + a3));
        // prefetch next step's pre row (latency hidden behind gates+barriers)
        if (t + 1 < S) p = pre[(size_t)(t + 1) * 256 + j];
        g_sh[j] = g;
        __syncthreads();
        if (j < 64) {
            float ig = fast_sigmoid(g_sh[j]);
            float fg = fast_sigmoid(g_sh[j + 64]);
            float gc = fast_tanh(g_sh[j + 128]);
            float og = fast_sigmoid(g_sh[j + 192]);
            c = fg * c + ig * gc;
            float h = og * fast_tanh(c);
            h_sh[j] = h;
            htrace[(size_t)t * 64 + j] = h;
        }
        __syncthreads();
    }
}

// ---------------------------------------------------------------------------
// Kernel 3/5: C[S,N] = A[S,64] @ W[N,64]^T + bias  via V_WMMA_F32_16X16X4_F32.
// One M-strip (16 rows x all N) per wave:
//   - A fragments for K=0..63 loaded ONCE into 32 VGPRs, reused for all N-tiles
//   - W (<=64KB) stays resident in L2/WGP$ across the grid
//   - A row index clamped (branch-free): garbage rows only feed unstored outputs
//   - all 16 B fragments batched before the 16-WMMA accumulation chain
// Fragment layouts per CDNA5 ISA 7.12.2:
//   A 16x4: lanes 0-15 hold K={k0,k0+1}, lanes 16-31 hold K={k0+2,k0+3}
//   B 4x16: mirrored (row K striped across lanes within a VGPR)
//   C/D:    VGPR v -> M = v + 8*(lane>=16), N = lane&15
// ---------------------------------------------------------------------------
__global__ __launch_bounds__(128) void gemm_k64_wmma_kernel(
    const float* __restrict__ A,   // [S,64]
    const float* __restrict__ W,   // [N,64] row-major (B = W^T)
    const float* __restrict__ bias0,
    const float* __restrict__ bias1,  // may be null
    float* __restrict__ C,         // [S,N]
    int S, int N)
{
    const int lane = threadIdx.x;       // 0..31
    const int wave = threadIdx.y;       // 0..3
    const long mtiles = ((long)S + 15) >> 4;
    const long mt = (long)blockIdx.x * 4 + wave;
    if (mt >= mtiles) return;

    const int m0 = (int)(mt << 4);
    const int lh = lane & 15;
    const int kh = (lane >> 4) << 1;   // 0 for lanes 0-15, 2 for lanes 16-31

    // Clamp A row (branch-free): out-of-range rows read valid data whose
    // products land only in C rows that are never stored.
    int arow = m0 + lh;
    arow = (arow < S) ? arow : (S - 1);
    const float* Arow = A + (size_t)arow * 64 + kh;

    // Load all 16 A fragments (K=0..63) once
    v2f a[16];
#pragma unroll
    for (int k = 0; k < 16; ++k) {
        a[k].x = Arow[4 * k];
        a[k].y = Arow[4 * k + 1];
    }

    const int mbase = m0 + ((lane >> 4) << 3);
    const int ntiles = N >> 4;

    for (int nt = 0; nt < ntiles; ++nt) {
        const int n0 = nt << 4;
        const float* Wrow = W + (size_t)(n0 + lh) * 64 + kh;

        // Batch all 16 B fragments so loads issue back-to-back
        v2f bf[16];
#pragma unroll
        for (int k = 0; k < 16; ++k) {
            bf[k].x = Wrow[4 * k];
            bf[k].y = Wrow[4 * k + 1];
        }

        v8f acc = {};
#pragma unroll
        for (int k = 0; k < 16; ++k) {
            acc = __builtin_amdgcn_wmma_f32_16x16x4_f32(
                /*neg_a=*/false, a[k], /*neg_b=*/false, bf[k],
                /*c_mod=*/(short)0, acc, /*reuse_a=*/false, /*reuse_b=*/false);
        }

        float bias = bias0[n0 + lh] + (bias1 ? bias1[n0 + lh] : 0.0f);
#pragma unroll
        for (int v = 0; v < 8; ++v) {
            int row = mbase + v;
            if (row < S) C[(size_t)row * N + (n0 + lh)] = acc[v] + bias;
        }
    }
}

// ---------------------------------------------------------------------------
// Launch: pre0 -> scan0 -> WMMA gemm (pre1, reusing pre buffer) -> scan1 ->
//         WMMA gemm (output FC).
// ---------------------------------------------------------------------------
extern "C" void kernel_launch(void* const* d_in, const int* in_sizes, int n_in,
                              void* d_out, int out_size, void* d_ws, size_t ws_size,
                              hipStream_t stream) {
    const float* y     = (const float*)d_in[0];
    const float* W_ih0 = (const float*)d_in[1];
    const float* W_hh0 = (const float*)d_in[2];
    const float* b_ih0 = (const float*)d_in[3];
    const float* b_hh0 = (const float*)d_in[4];
    const float* W_ih1 = (const float*)d_in[5];
    const float* W_hh1 = (const float*)d_in[6];
    const float* b_ih1 = (const float*)d_in[7];
    const float* b_hh1 = (const float*)d_in[8];
    const float* W_fc  = (const float*)d_in[9];
    const float* b_fc  = (const float*)d_in[10];

    const int T = in_sizes[0];
    const int S = T - 1;
    float* out = (float*)d_out;

    // workspace: pre [S,256] (reused for both layers) | h0 [S,64] | h1 [S,64]
    float* pre = (float*)d_ws;
    float* h0  = pre + (size_t)S * 256;
    float* h1  = h0  + (size_t)S * 64;

    const long mtiles = ((long)S + 15) / 16;
    const unsigned gemm_blocks = (unsigned)((mtiles + 3) / 4);
    const dim3 gemm_blk(32, 4);

    // 1) input contribution for layer 0
    pre0_kernel<<<S, 256, 0, stream>>>(y, W_ih0, b_ih0, b_hh0, pre, S);

    // 2) serial scan, layer 0
    lstm_scan_kernel<<<1, 256, 0, stream>>>(pre, W_hh0, h0, S);

    // 3) input contribution for layer 1 via WMMA GEMM (overwrites pre)
    gemm_k64_wmma_kernel<<<gemm_blocks, gemm_blk, 0, stream>>>(
        h0, W_ih1, b_ih1, b_hh1, pre, S, 256);

    // 4) serial scan, layer 1
    lstm_scan_kernel<<<1, 256, 0, stream>>>(pre, W_hh1, h1, S);

    // 5) output FC via WMMA GEMM
    gemm_k64_wmma_kernel<<<gemm_blocks, gemm_blk, 0, stream>>>(
        h1, W_fc, b_fc, nullptr, out, S, 64);
}